// SimpleBlockSingleGraph_4054449127562
// MI455X (gfx1250) — compile-verified
//
#include <hip/hip_runtime.h>

typedef __attribute__((ext_vector_type(2))) float v2f;
typedef __attribute__((ext_vector_type(8))) float v8f;

#define N_NODES   204800      // BATCH*WINDOW*NUM_SERIES
#define C_DIM     64
#define N_EDGES   2000000
#define NSERIES   50
#define WINDOW_T  128
#define NEG_SLOPE 0.01f

// ---------------------------------------------------------------- h = b_gcn
__global__ void init_h_kernel(const float* __restrict__ b_gcn, float* __restrict__ h) {
    long long i = (long long)blockIdx.x * blockDim.x + threadIdx.x;
    if (i < (long long)N_NODES * C_DIM) h[i] = b_gcn[i & (C_DIM - 1)];
}

// ------------------------------------------------- deg[col[e]] += E[e]
__global__ void deg_kernel(const int* __restrict__ col, const float* __restrict__ E,
                           float* __restrict__ deg) {
    int e = blockIdx.x * blockDim.x + threadIdx.x;
    if (e < N_EDGES) atomicAdd(&deg[col[e]], E[e]);
}

// -------------------------------------- dinv = deg>0 ? deg^-1/2 : 0 (in place)
__global__ void dinv_kernel(float* __restrict__ deg) {
    int i = blockIdx.x * blockDim.x + threadIdx.x;
    if (i < N_NODES) {
        float d = deg[i];
        deg[i] = (d > 0.f) ? rsqrtf(fmaxf(d, 1e-30f)) : 0.f;
    }
}

// ------------------------------ norm[e] = dinv[row]*E*dinv[col]
__global__ void norm_kernel(const int* __restrict__ row, const int* __restrict__ col,
                            const float* __restrict__ E, const float* __restrict__ dinv,
                            float* __restrict__ normv) {
    int e = blockIdx.x * blockDim.x + threadIdx.x;
    if (e < N_EDGES) normv[e] = dinv[row[e]] * E[e] * dinv[col[e]];
}

// ------------------------------------------- xw = x @ W_gcn  (fp32 WMMA)
// One wave -> 16x16 output tile. grid.x = N/16 row tiles, 4 waves cover 64 cols.
__global__ __launch_bounds__(128) void xw_wmma_kernel(const float* __restrict__ x,
                                                      const float* __restrict__ Wg,
                                                      float* __restrict__ xw) {
    const int lane = threadIdx.x & 31;
    const int wave = threadIdx.x >> 5;
    const int m    = lane & 15;       // A row within tile (both lane halves)
    const int hi   = lane >> 4;       // lane half selects K pair
    const long long arow = (long long)blockIdx.x * 16 + m;
    const int n    = wave * 16 + m;   // output column
    v8f acc = {};
#pragma unroll
    for (int kb = 0; kb < 64; kb += 4) {
        const int k0 = kb + 2 * hi;
        v2f a, b;
        a.x = x[arow * C_DIM + k0];
        a.y = x[arow * C_DIM + k0 + 1];
        b.x = Wg[k0 * C_DIM + n];           // W_gcn[k][n], row-major (C_IN, C_MID)
        b.y = Wg[(k0 + 1) * C_DIM + n];
        acc = __builtin_amdgcn_wmma_f32_16x16x4_f32(false, a, false, b,
                                                    (short)0, acc, false, false);
    }
    const long long rbase = (long long)blockIdx.x * 16 + 8 * hi;
#pragma unroll
    for (int v = 0; v < 8; ++v)
        xw[(rbase + v) * C_DIM + n] = acc[v];
}

// ---------------------- h[col] += xw[row] * norm  (64 threads per edge, L2-resident)
__global__ void scatter_kernel(const int* __restrict__ row, const int* __restrict__ col,
                               const float* __restrict__ normv,
                               const float* __restrict__ xw, float* __restrict__ h) {
    long long i = (long long)blockIdx.x * blockDim.x + threadIdx.x;
    if (i < (long long)N_EDGES * C_DIM) {
        const int e = (int)(i >> 6);
        const int c = (int)(i & 63);
        const int r  = row[e];
        const int cl = col[e];
        atomicAdd(&h[(long long)cl * C_DIM + c], xw[(long long)r * C_DIM + c] * normv[e]);
    }
}

// ------------- Conv1d(k=3, same-pad) + bias + LeakyReLU as 3 shifted WMMA GEMMs
// Node index n=(b*T+t)*Ns+s: tap k reads row n+(k-1)*Ns, valid iff 0<=t+k-1<T.
__global__ __launch_bounds__(128) void conv_wmma_kernel(const float* __restrict__ h,
                                                        const float* __restrict__ Wc,
                                                        const float* __restrict__ bias,
                                                        float* __restrict__ out) {
    const int lane = threadIdx.x & 31;
    const int wave = threadIdx.x >> 5;
    const int m    = lane & 15;
    const int hi   = lane >> 4;
    const int nrow = blockIdx.x * 16 + m;
    const int t    = (nrow / NSERIES) % WINDOW_T;
    const int co   = wave * 16 + m;
    v8f acc = {};
#pragma unroll
    for (int tap = 0; tap < 3; ++tap) {
        const int  tt    = t + tap - 1;
        const bool valid = ((unsigned)tt < (unsigned)WINDOW_T);
        const long long src = (long long)(nrow + (tap - 1) * NSERIES) * C_DIM;
#pragma unroll
        for (int kb = 0; kb < 64; kb += 4) {
            const int k0 = kb + 2 * hi;
            v2f a, b;
            a.x = valid ? h[src + k0]     : 0.f;   // zero-pad in time
            a.y = valid ? h[src + k0 + 1] : 0.f;
            b.x = Wc[co * (C_DIM * 3) + k0 * 3 + tap];        // W_conv[co][ci][k]
            b.y = Wc[co * (C_DIM * 3) + (k0 + 1) * 3 + tap];
            acc = __builtin_amdgcn_wmma_f32_16x16x4_f32(false, a, false, b,
                                                        (short)0, acc, false, false);
        }
    }
    const long long rbase = (long long)blockIdx.x * 16 + 8 * hi;
    const float bc = bias[co];
#pragma unroll
    for (int v = 0; v < 8; ++v) {
        float val = acc[v] + bc;
        out[(rbase + v) * C_DIM + co] = (val >= 0.f) ? val : NEG_SLOPE * val;
    }
}

extern "C" void kernel_launch(void* const* d_in, const int* in_sizes, int n_in,
                              void* d_out, int out_size, void* d_ws, size_t ws_size,
                              hipStream_t stream) {
    const float* x      = (const float*)d_in[0];
    const int*   A      = (const int*)  d_in[1];   // (2, NUM_EDGES) flat
    const float* E      = (const float*)d_in[2];
    const float* W_gcn  = (const float*)d_in[3];
    const float* b_gcn  = (const float*)d_in[4];
    const float* W_conv = (const float*)d_in[5];
    const float* b_conv = (const float*)d_in[6];
    float*       out    = (float*)d_out;

    const int* row = A;
    const int* col = A + N_EDGES;

    // Workspace layout (floats): deg/dinv | norm | xw | h   (~114 MB total)
    float* deg   = (float*)d_ws;
    float* normv = deg + N_NODES;
    float* xw    = normv + N_EDGES;
    float* h     = xw + (size_t)N_NODES * C_DIM;

    hipMemsetAsync(deg, 0, (size_t)N_NODES * sizeof(float), stream);

    {   // h := b_gcn broadcast (segment_sum accumulates on top)
        long long tot = (long long)N_NODES * C_DIM;
        init_h_kernel<<<(unsigned)((tot + 255) / 256), 256, 0, stream>>>(b_gcn, h);
    }
    deg_kernel <<<(N_EDGES + 255) / 256, 256, 0, stream>>>(col, E, deg);
    dinv_kernel<<<(N_NODES + 255) / 256, 256, 0, stream>>>(deg);
    norm_kernel<<<(N_EDGES + 255) / 256, 256, 0, stream>>>(row, col, E, deg, normv);

    xw_wmma_kernel<<<N_NODES / 16, 128, 0, stream>>>(x, W_gcn, xw);

    {   // 2M edges * 64 channels, one atomic each
        long long tot = (long long)N_EDGES * C_DIM;
        scatter_kernel<<<(unsigned)((tot + 255) / 256), 256, 0, stream>>>(row, col, normv, xw, h);
    }

    conv_wmma_kernel<<<N_NODES / 16, 128, 0, stream>>>(h, W_conv, b_conv, out);
}